// Mistral3PatchMerger_8856222564596
// MI455X (gfx1250) — compile-verified
//
#include <hip/hip_runtime.h>

typedef __attribute__((ext_vector_type(2))) float v2f;
typedef __attribute__((ext_vector_type(8))) float v8f;

#define PATCH  14
#define HIDDEN 1024
#define NIMG   8
#define KTOT   4096
#define BM     64
#define BN     128
#define BK     32
#define NT     (KTOT / BK)   // 128 K-tiles
#define ASTR   36            // A row stride (32+4 pad floats)
#define TSTR   36            // B^T column stride (32+4 pad floats)

__global__ __launch_bounds__(256)
void Mistral3PatchMerger_gemm(const float* __restrict__ feats,
                              const int*   __restrict__ sizes,
                              const float* __restrict__ Wm,
                              float*       __restrict__ out,
                              int Mtot) {
  __shared__ float a_lds[2][BM * ASTR];    // 2 x 9216 B, row-major A tile
  __shared__ float bt_lds[2][BN * TSTR];   // 2 x 18432 B, column-major W tile
  __shared__ int   s_base[BM];
  __shared__ int   s_w[BM];

  const int tid = threadIdx.x;
  const int m0  = blockIdx.y * BM;
  const int n0  = blockIdx.x * BN;

  // ---- merged row m -> base token index + patch-grid width (gather mapping)
  if (tid < BM) {
    int m = m0 + tid;
    if (m >= Mtot) m = Mtot - 1;             // clamp; stores masked later
    int tok_off = 0, mer_off = 0, base = 0, wgrid = 2;
    for (int i = 0; i < NIMG; ++i) {
      int hg = sizes[2 * i]     / PATCH;
      int wg = sizes[2 * i + 1] / PATCH;
      int nmer = (hg >> 1) * (wg >> 1);
      if (m >= mer_off && m < mer_off + nmer) {
        int local = m - mer_off;
        int r = local / (wg >> 1);
        int c = local - r * (wg >> 1);
        base  = tok_off + (2 * r) * wg + 2 * c;
        wgrid = wg;
      }
      tok_off += hg * wg;
      mer_off += nmer;
    }
    s_base[tid] = base;
    s_w[tid]    = wgrid;
  }
  __syncthreads();

  // ---- per-thread load coordinates
  const int arow0  = tid >> 3;              // A: 8 float4 per 32-float row
  const int af4    = tid & 7;
  const int abase0 = s_base[arow0],      aw0 = s_w[arow0];
  const int abase1 = s_base[arow0 + 32], aw1 = s_w[arow0 + 32];

  const int bcol = tid & 127;               // one W column per thread
  const int brg0 = tid >> 7;                // row-group 0..1; +2*j per pass

  const int lane = tid & 31;
  const int wid  = tid >> 5;
  const int wm   = wid >> 2;                // 0..1 -> 32-row strip
  const int wn   = wid & 3;                 // 0..3 -> 32-col strip
  const int half = lane >> 4;
  const int l16  = lane & 15;

  v8f acc[2][2] = {};

  // register staging for double buffer
  float4 aReg0, aReg1;
  float4 bReg[4];

  auto load_tile = [&](int k0) {
    const int chunk    = k0 >> 10;          // which of the 4 merged source tokens
    const int dy       = chunk >> 1, dx = chunk & 1;
    const int featbase = k0 & 1023;
    long tok0 = (long)(abase0 + dy * aw0 + dx);
    long tok1 = (long)(abase1 + dy * aw1 + dx);
    aReg0 = *(const float4*)(feats + tok0 * HIDDEN + featbase + af4 * 4);
    aReg1 = *(const float4*)(feats + tok1 * HIDDEN + featbase + af4 * 4);
#pragma unroll
    for (int j = 0; j < 4; ++j) {
      int rg = brg0 + 2 * j;                // 0..7 -> rows 4*rg..4*rg+3
      const float* src = Wm + (size_t)(k0 + 4 * rg) * HIDDEN + n0 + bcol;
      bReg[j].x = src[0 * HIDDEN];          // coalesced b32 (W is L2-resident)
      bReg[j].y = src[1 * HIDDEN];
      bReg[j].z = src[2 * HIDDEN];
      bReg[j].w = src[3 * HIDDEN];
    }
  };

  auto store_tile = [&](int buf) {
    *(float4*)&a_lds[buf][ arow0       * ASTR + af4 * 4] = aReg0;
    *(float4*)&a_lds[buf][(arow0 + 32) * ASTR + af4 * 4] = aReg1;
#pragma unroll
    for (int j = 0; j < 4; ++j) {
      int rg = brg0 + 2 * j;
      *(float4*)&bt_lds[buf][bcol * TSTR + 4 * rg] = bReg[j];  // transposed store
    }
  };

  load_tile(0);
  for (int t = 0; t < NT; ++t) {
    const int buf = t & 1;
    store_tile(buf);
    if (t + 1 < NT) load_tile((t + 1) * BK);  // prefetch next tile into regs
    __syncthreads();                           // single barrier per K-tile

    const float* aB = a_lds[buf];
    const float* bB = bt_lds[buf];
#pragma unroll
    for (int kk = 0; kk < BK; kk += 4) {
      v2f afrag[2], bfrag[2];
#pragma unroll
      for (int sm = 0; sm < 2; ++sm)
        afrag[sm] = *(const v2f*)&aB[(32 * wm + 16 * sm + l16) * ASTR + kk + 2 * half];
#pragma unroll
      for (int sn = 0; sn < 2; ++sn)
        bfrag[sn] = *(const v2f*)&bB[(32 * wn + 16 * sn + l16) * TSTR + kk + 2 * half];
#pragma unroll
      for (int sm = 0; sm < 2; ++sm)
#pragma unroll
        for (int sn = 0; sn < 2; ++sn)
          acc[sm][sn] = __builtin_amdgcn_wmma_f32_16x16x4_f32(
              false, afrag[sm], false, bfrag[sn],
              (short)0, acc[sm][sn], false, false);
    }
  }

  // ---- store C tiles: VGPR v -> row v + 8*half, col = l16 (ISA C/D layout)
#pragma unroll
  for (int sm = 0; sm < 2; ++sm) {
#pragma unroll
    for (int sn = 0; sn < 2; ++sn) {
      int colg = n0 + 32 * wn + 16 * sn + l16;
#pragma unroll
      for (int v = 0; v < 8; ++v) {
        int rowg = m0 + 32 * wm + 16 * sm + 8 * half + v;
        if (rowg < Mtot)
          out[(size_t)rowg * HIDDEN + colg] = acc[sm][sn][v];
      }
    }
  }
}

extern "C" void kernel_launch(void* const* d_in, const int* in_sizes, int n_in,
                              void* d_out, int out_size, void* d_ws, size_t ws_size,
                              hipStream_t stream) {
  const float* feats = (const float*)d_in[0];   // [59808, 1024] fp32
  const int*   sizes = (const int*)  d_in[1];   // [8, 2] int32 pixel sizes
  const float* Wm    = (const float*)d_in[2];   // [4096, 1024] fp32
  float*       out   = (float*)d_out;           // [14952, 1024] fp32
  (void)d_ws; (void)ws_size; (void)n_in; (void)in_sizes;

  int Mtot = out_size / HIDDEN;                 // 14952
  dim3 grid(HIDDEN / BN, (Mtot + BM - 1) / BM); // (8, 234)
  Mistral3PatchMerger_gemm<<<grid, 256, 0, stream>>>(feats, sizes, Wm, out, Mtot);
}